// NeuralNet_19250043421419
// MI455X (gfx1250) — compile-verified
//
#include <hip/hip_runtime.h>
#include <hip/hip_bf16.h>

typedef float v2f __attribute__((ext_vector_type(2)));
typedef float v4f __attribute__((ext_vector_type(4)));
typedef float v8f __attribute__((ext_vector_type(8)));

#define DIN 3072
#define NEG_SLOPE 0.01f

__device__ __forceinline__ float leaky(float v) {
    return v >= 0.0f ? v : NEG_SLOPE * v;
}

// ---------------------------------------------------------------------------
// Kernel 1: per-output-column sums of w1 (needed for the fused normalization
// fix-up).  w1 is (32, 3072) row-major; one block per j.
// ---------------------------------------------------------------------------
__global__ __launch_bounds__(256) void w1sum_kernel(const float* __restrict__ w1,
                                                    float* __restrict__ ws) {
    __shared__ float red[256];
    const int j = blockIdx.x;
    const float* row = w1 + (size_t)j * DIN;
    float s = 0.0f;
    for (int k = threadIdx.x; k < DIN; k += 256) s += row[k];
    red[threadIdx.x] = s;
    __syncthreads();
    for (int off = 128; off > 0; off >>= 1) {
        if (threadIdx.x < off) red[threadIdx.x] += red[threadIdx.x + off];
        __syncthreads();
    }
    if (threadIdx.x == 0) ws[j] = red[0];
}

// ---------------------------------------------------------------------------
// Kernel 2: fused  rownorm -> (x@w1^T+b1) -> leaky -> (..@w2^T+b2) -> leaky
//                  -> (..@w3^T+b3) -> leaky
// One wave handles 16 rows; 8 waves / block -> 128 rows / block.
// Single streaming pass over x using V_WMMA_F32_16X16X4_F32.
// x is read-once -> non-temporal loads + WGP-scope prefetch ahead of the
// stream; w1/w2/w3 stay L2/L0 resident.
// ---------------------------------------------------------------------------
__global__ __launch_bounds__(256) void mlp_kernel(
    const float* __restrict__ x,
    const float* __restrict__ w1, const float* __restrict__ b1,
    const float* __restrict__ w2, const float* __restrict__ b2,
    const float* __restrict__ w3, const float* __restrict__ b3,
    const float* __restrict__ w1sum,
    float* __restrict__ out)
{
    __shared__ float meanv[8][16];
    __shared__ float invv[8][16];
    __shared__ float h1s[8][16][33];   // padded to dodge bank conflicts
    __shared__ float h2s[8][16][33];

    const int lane = threadIdx.x & 31;
    const int wv   = threadIdx.x >> 5;           // wave id in block (0..7)
    const int r    = lane & 15;                  // row-in-tile / N column
    const int h    = lane >> 4;                  // lane half (K split)
    const int rb   = (blockIdx.x * 8 + wv) * 16; // base row of this wave

    v8f c0 = {};   // columns N = 0..15
    v8f c1 = {};   // columns N = 16..31
    float s = 0.0f, ss = 0.0f;

    const float* xrow  = x  + (size_t)(rb + r) * DIN + 4 * h;
    const float* w1p0  = w1 + (size_t)r        * DIN + 2 * h;   // n = r
    const float* w1p1  = w1 + (size_t)(r + 16) * DIN + 2 * h;   // n = r+16

    for (int kb = 0; kb < DIN; kb += 8) {
        // read-once stream: non-temporal b128 load + near-cache prefetch ahead
        v4f f = __builtin_nontemporal_load((const v4f*)(xrow + kb));
        __builtin_prefetch(xrow + kb + 256, 0, 3);

        // running row sum / sum-of-squares (this lane covers K = kb+4h..+3)
        s  += f.x + f.y + f.z + f.w;
        ss  = fmaf(f.x, f.x, fmaf(f.y, f.y, fmaf(f.z, f.z, fmaf(f.w, f.w, ss))));

        // redistribute the float4 between lane halves into two K=4 A-tiles:
        // tile0 (K=kb..kb+3): h=0 -> own .xy ; h=1 -> partner .zw
        // tile1 (K=kb+4..+7): h=0 -> partner .xy ; h=1 -> own .zw
        float gx = __shfl_xor(f.x, 16, 32);
        float gy = __shfl_xor(f.y, 16, 32);
        float gz = __shfl_xor(f.z, 16, 32);
        float gw = __shfl_xor(f.w, 16, 32);
        v2f a0, a1;
        a0.x = h ? gz : f.x;  a0.y = h ? gw : f.y;
        a1.x = h ? f.z : gx;  a1.y = h ? f.w : gy;

        // B tiles: lane(r,h) holds w1[n][k+2h], w1[n][k+2h+1]  (L2/L0-hot)
        v2f b00 = *(const v2f*)(w1p0 + kb);
        v2f b01 = *(const v2f*)(w1p1 + kb);
        v2f b10 = *(const v2f*)(w1p0 + kb + 4);
        v2f b11 = *(const v2f*)(w1p1 + kb + 4);

        c0 = __builtin_amdgcn_wmma_f32_16x16x4_f32(false, a0, false, b00,
                                                   (short)0, c0, false, false);
        c1 = __builtin_amdgcn_wmma_f32_16x16x4_f32(false, a0, false, b01,
                                                   (short)0, c1, false, false);
        c0 = __builtin_amdgcn_wmma_f32_16x16x4_f32(false, a1, false, b10,
                                                   (short)0, c0, false, false);
        c1 = __builtin_amdgcn_wmma_f32_16x16x4_f32(false, a1, false, b11,
                                                   (short)0, c1, false, false);
    }

    // finish per-row statistics (combine the two K lane-halves)
    s  += __shfl_xor(s, 16, 32);
    ss += __shfl_xor(ss, 16, 32);
    const float mean   = s * (1.0f / (float)DIN);
    const float var    = (ss - s * s * (1.0f / (float)DIN)) * (1.0f / (float)(DIN - 1));
    const float invstd = 1.0f / sqrtf(var);
    if (lane < 16) { meanv[wv][r] = mean; invv[wv][r] = invstd; }
    __syncthreads();

    // fold normalization into the accumulated raw dots, add bias, leaky, ->LDS
    const float ws0 = w1sum[r], ws1 = w1sum[r + 16];
    const float bb0 = b1[r],    bb1 = b1[r + 16];
#pragma unroll
    for (int v = 0; v < 8; ++v) {
        const int M = v + 8 * h;                 // row within 16-row tile
        const float m  = meanv[wv][M];
        const float is = invv[wv][M];
        h1s[wv][M][r]      = leaky((c0[v] - m * ws0) * is + bb0);
        h1s[wv][M][r + 16] = leaky((c1[v] - m * ws1) * is + bb1);
    }
    __syncthreads();

    // layer 2: h2[row, lane] = leaky(b2[lane] + sum_j h1[row,j]*w2[lane,j])
    float w2row[32];
#pragma unroll
    for (int j = 0; j < 32; ++j) w2row[j] = w2[lane * 32 + j];
    const float bb2 = b2[lane];
    for (int row = 0; row < 16; ++row) {
        float acc = bb2;
#pragma unroll
        for (int j = 0; j < 32; ++j) acc = fmaf(h1s[wv][row][j], w2row[j], acc);
        h2s[wv][row][lane] = leaky(acc);
    }
    __syncthreads();

    // layer 3: 16 rows x 10 outs = 160 = 5*32 -> fully converged loop
#pragma unroll
    for (int t = 0; t < 5; ++t) {
        const int idx = t * 32 + lane;
        const int row = idx / 10;
        const int n   = idx - row * 10;
        float acc = b3[n];
#pragma unroll
        for (int j = 0; j < 32; ++j) acc = fmaf(h2s[wv][row][j], w3[n * 32 + j], acc);
        __builtin_nontemporal_store(leaky(acc), out + (size_t)(rb + row) * 10 + n);
    }
}

extern "C" void kernel_launch(void* const* d_in, const int* in_sizes, int n_in,
                              void* d_out, int out_size, void* d_ws, size_t ws_size,
                              hipStream_t stream) {
    const float* x  = (const float*)d_in[0];
    const float* w1 = (const float*)d_in[1];
    const float* b1 = (const float*)d_in[2];
    const float* w2 = (const float*)d_in[3];
    const float* b2 = (const float*)d_in[4];
    const float* w3 = (const float*)d_in[5];
    const float* b3 = (const float*)d_in[6];
    float* out   = (float*)d_out;
    float* w1sum = (float*)d_ws;   // 32 floats of scratch

    w1sum_kernel<<<32, 256, 0, stream>>>(w1, w1sum);
    // 32768 rows / (8 waves * 16 rows) = 256 blocks
    mlp_kernel<<<256, 256, 0, stream>>>(x, w1, b1, w2, b2, w3, b3, w1sum, out);
}